// GraphConvolutionBlock_3770981286190
// MI455X (gfx1250) — compile-verified
//
#include <hip/hip_runtime.h>

typedef float v2f __attribute__((ext_vector_type(2)));
typedef float v8f __attribute__((ext_vector_type(8)));

#define N_NODES 50000
#define F_DIM   128
#define H_DIM   128
#define N_EDGES 800000
#define BN_EPS  1e-5f

// ---------------------------------------------------------------------------
// Zero the BN statistics accumulators (sum[128], sumsq[128]).
// ---------------------------------------------------------------------------
__global__ void k_init_stats(float* __restrict__ sums, float* __restrict__ sumsq) {
    int i = threadIdx.x;
    if (i < H_DIM) { sums[i] = 0.0f; sumsq[i] = 0.0f; }
}

// ---------------------------------------------------------------------------
// agg[n,h] = bias[h]   (fuses the "+ bias" of the reference into the
// scatter-add accumulator's initial value)
// ---------------------------------------------------------------------------
__global__ void k_init_agg(float* __restrict__ agg, const float* __restrict__ bias) {
    int idx = blockIdx.x * blockDim.x + threadIdx.x;
    if (idx < N_NODES * H_DIM) agg[idx] = bias[idx & (H_DIM - 1)];
}

// ---------------------------------------------------------------------------
// xw = feature @ weight via V_WMMA_F32_16X16X4_F32 (exact fp32).
// blockDim = 256 (8 waves). blockIdx.x = 16-row tile (N = 3125*16 exactly).
// wave id  = 16-col tile   (H = 8*16 exactly).  No tails anywhere.
//
// fp32 A (16x4) fragment, ISA 7.12.2: lanes 0-15 hold K=k,k+1 in v[0],v[1];
// lanes 16-31 hold K=k+2,k+3.  B (4x16): row striped across lanes per VGPR,
// lower half-wave rows k,k+1 / upper half rows k+2,k+3.  C/D: VGPR i holds
// row i (lanes 0-15) and row i+8 (lanes 16-31).
// ---------------------------------------------------------------------------
__global__ void k_gemm(const float* __restrict__ A,   // [N, F] row-major
                       const float* __restrict__ B,   // [F, H] row-major
                       float* __restrict__ C) {       // [N, H] row-major
    const int wave = threadIdx.x >> 5;   // 0..7 -> column tile
    const int lane = threadIdx.x & 31;
    const int half = lane >> 4;          // 0 or 1
    const int l16  = lane & 15;
    const int m0   = blockIdx.x * 16;
    const int n0   = wave * 16;

    v8f acc = {0.f, 0.f, 0.f, 0.f, 0.f, 0.f, 0.f, 0.f};
    const float* Arow = A + (m0 + l16) * F_DIM;
    const float* Bcol = B + n0 + l16;

    #pragma unroll
    for (int k = 0; k < F_DIM; k += 4) {
        const int ka = k + 2 * half;
        v2f a;
        a.x = Arow[ka];
        a.y = Arow[ka + 1];
        v2f b;
        b.x = Bcol[ka * H_DIM];
        b.y = Bcol[(ka + 1) * H_DIM];
        acc = __builtin_amdgcn_wmma_f32_16x16x4_f32(
                  /*neg_a=*/false, a, /*neg_b=*/false, b,
                  /*c_mod=*/(short)0, acc, /*reuse_a=*/false, /*reuse_b=*/false);
    }

    #pragma unroll
    for (int i = 0; i < 8; ++i) {
        C[(m0 + i + 8 * half) * H_DIM + n0 + l16] = acc[i];
    }
}

// ---------------------------------------------------------------------------
// COO SpMM scatter-add: agg[row] += w_e * xw[col].
// 32 consecutive threads own one edge; each thread handles a float4 slice of
// the 128-float row -> fully coalesced 512B gathers of xw rows (L2 resident)
// and coalesced f32 atomic adds.
// ---------------------------------------------------------------------------
__global__ void k_spmm(const float* __restrict__ xw,
                       const float* __restrict__ ew,
                       const int*   __restrict__ er,
                       const int*   __restrict__ ec,
                       float* __restrict__ agg) {
    int idx = blockIdx.x * blockDim.x + threadIdx.x;     // E*32 = 25.6M threads
    if (idx >= N_EDGES * 32) return;
    const int e  = idx >> 5;
    const int h4 = (idx & 31) << 2;                      // 0,4,...,124
    const int r  = er[e];
    const int c  = ec[e];
    const float w = ew[e];

    const float4 v = *(const float4*)(xw + c * H_DIM + h4);
    float* dst = agg + r * H_DIM + h4;
    atomicAdd(dst + 0, w * v.x);
    atomicAdd(dst + 1, w * v.y);
    atomicAdd(dst + 2, w * v.z);
    atomicAdd(dst + 3, w * v.w);
}

// ---------------------------------------------------------------------------
// Column-wise sum / sum-of-squares over agg. Each 128-thread block owns all
// columns, strides the row dimension, accumulates in registers, then one
// atomic per column per block.
// ---------------------------------------------------------------------------
__global__ void k_stats(const float* __restrict__ agg,
                        float* __restrict__ sums,
                        float* __restrict__ sumsq) {
    const int h = threadIdx.x;                           // 0..127
    float s = 0.0f, s2 = 0.0f;
    for (int r = blockIdx.x; r < N_NODES; r += gridDim.x) {
        const float v = agg[r * H_DIM + h];
        s  += v;
        s2 += v * v;
    }
    atomicAdd(&sums[h], s);
    atomicAdd(&sumsq[h], s2);
}

// ---------------------------------------------------------------------------
// out = feature + relu( (agg - mean) * rsqrt(var + eps) * gamma + beta )
// ---------------------------------------------------------------------------
__global__ void k_final(const float* __restrict__ agg,
                        const float* __restrict__ feat,
                        const float* __restrict__ sums,
                        const float* __restrict__ sumsq,
                        const float* __restrict__ gamma,
                        const float* __restrict__ beta,
                        float* __restrict__ out) {
    int idx = blockIdx.x * blockDim.x + threadIdx.x;
    if (idx >= N_NODES * H_DIM) return;
    const int h = idx & (H_DIM - 1);
    const float inv_n = 1.0f / (float)N_NODES;
    const float mean  = sums[h] * inv_n;
    const float var   = sumsq[h] * inv_n - mean * mean;
    const float scale = rsqrtf(var + BN_EPS) * gamma[h];
    float v = (agg[idx] - mean) * scale + beta[h];
    v = v > 0.0f ? v : 0.0f;
    out[idx] = feat[idx] + v;
}

// ---------------------------------------------------------------------------
// Inputs (setup_inputs order):
//  0 feature [N,F] f32   1 weight [F,H] f32   2 bias [H] f32
//  3 bn_gamma [H] f32    4 bn_beta [H] f32    5 edge_weight [E] f32
//  6 edge_row [E] i32    7 edge_col [E] i32
// Output: [N,H] f32.
// Workspace: xw (N*H f32) | agg (N*H f32) | sums (H) | sumsq (H)
// ---------------------------------------------------------------------------
extern "C" void kernel_launch(void* const* d_in, const int* in_sizes, int n_in,
                              void* d_out, int out_size, void* d_ws, size_t ws_size,
                              hipStream_t stream) {
    const float* feature = (const float*)d_in[0];
    const float* weight  = (const float*)d_in[1];
    const float* bias    = (const float*)d_in[2];
    const float* gamma   = (const float*)d_in[3];
    const float* beta    = (const float*)d_in[4];
    const float* ew      = (const float*)d_in[5];
    const int*   er      = (const int*)d_in[6];
    const int*   ec      = (const int*)d_in[7];
    float*       out     = (float*)d_out;

    float* xw    = (float*)d_ws;
    float* agg   = xw + N_NODES * H_DIM;
    float* sums  = agg + N_NODES * H_DIM;
    float* sumsq = sums + H_DIM;

    const int nh_blocks = (N_NODES * H_DIM + 255) / 256;

    k_init_stats<<<1, 256, 0, stream>>>(sums, sumsq);
    k_init_agg<<<nh_blocks, 256, 0, stream>>>(agg, bias);
    k_gemm<<<N_NODES / 16, 256, 0, stream>>>(feature, weight, xw);
    k_spmm<<<(N_EDGES * 32 + 255) / 256, 256, 0, stream>>>(xw, ew, er, ec, agg);
    k_stats<<<512, 128, 0, stream>>>(agg, sums, sumsq);
    k_final<<<nh_blocks, 256, 0, stream>>>(agg, feature, sums, sumsq, gamma, beta, out);
}